// LogitsModel_9586367004839
// MI455X (gfx1250) — compile-verified
//
#include <hip/hip_runtime.h>
#include <hip/hip_bf16.h>
#include <stdint.h>

// Problem constants (from reference)
#define BATCH   128
#define HIST    200
#define VOCAB   151936
#define MASKW   (VOCAB / 32)          // 4748 words per row
#define TOPK    50
#define TOPP    0.8f
#define PEN     1.1f

// Streaming / selection constants
#define TILE    1024                   // floats per block-tile (4 KB)
#define NMAIN   151552                 // 148 full tiles; 384-element tail
#define HBINS   4096                   // 12-bit key histogram
#define CAP     2048                   // candidate capacity (power of 2, bitonic)
#define NTHREADS 256                   // 8 wave32 per block

// ---------------------------------------------------------------------------
// Order-preserving float -> uint key (monotone increasing with float value)
// ---------------------------------------------------------------------------
__device__ __forceinline__ uint32_t f2key(float x) {
    uint32_t u = __float_as_uint(x);
    return (u & 0x80000000u) ? ~u : (u | 0x80000000u);
}
__device__ __forceinline__ float key2f(uint32_t k) {
    uint32_t u = (k & 0x80000000u) ? (k & 0x7FFFFFFFu) : ~k;
    return __uint_as_float(u);
}

// repetition penalty + key for one element
__device__ __forceinline__ uint32_t score_key(float x, const uint32_t* mrow, int idx) {
    uint32_t w = mrow[idx >> 5];
    if ((w >> (idx & 31)) & 1u) x = (x < 0.0f) ? x * PEN : x / PEN;
    return f2key(x);
}

// ---------------------------------------------------------------------------
// JAX threefry2x32(key=[0,42]) gumbel at flat index j of shape (B, V).
// random_bits counter layout: halves split; out[j] = x0 of (j, j+half) for
// j < half else x1 of (j-half, j).  bits -> [0,1) via (bits>>9)|1.0f trick.
// ---------------------------------------------------------------------------
__device__ __forceinline__ uint32_t rotl32(uint32_t x, int r) {
    return (x << r) | (x >> (32 - r));
}
__device__ float jax_gumbel42(uint32_t j) {
    const uint32_t k0 = 0u, k1 = 42u;
    const uint32_t ks2 = k0 ^ k1 ^ 0x1BD11BDAu;
    const uint32_t NHALF = (uint32_t)((unsigned long long)BATCH * VOCAB / 2ull); // 9723904
    uint32_t c0, c1; bool lo;
    if (j < NHALF) { c0 = j; c1 = j + NHALF; lo = true; }
    else           { c0 = j - NHALF; c1 = j; lo = false; }
    uint32_t x0 = c0 + k0, x1 = c1 + k1;
    x0 += x1; x1 = rotl32(x1, 13); x1 ^= x0;
    x0 += x1; x1 = rotl32(x1, 15); x1 ^= x0;
    x0 += x1; x1 = rotl32(x1, 26); x1 ^= x0;
    x0 += x1; x1 = rotl32(x1,  6); x1 ^= x0;
    x0 += k1; x1 += ks2 + 1u;
    x0 += x1; x1 = rotl32(x1, 17); x1 ^= x0;
    x0 += x1; x1 = rotl32(x1, 29); x1 ^= x0;
    x0 += x1; x1 = rotl32(x1, 16); x1 ^= x0;
    x0 += x1; x1 = rotl32(x1, 24); x1 ^= x0;
    x0 += ks2; x1 += k0 + 2u;
    x0 += x1; x1 = rotl32(x1, 13); x1 ^= x0;
    x0 += x1; x1 = rotl32(x1, 15); x1 ^= x0;
    x0 += x1; x1 = rotl32(x1, 26); x1 ^= x0;
    x0 += x1; x1 = rotl32(x1,  6); x1 ^= x0;
    x0 += k0; x1 += k1 + 3u;
    x0 += x1; x1 = rotl32(x1, 17); x1 ^= x0;
    x0 += x1; x1 = rotl32(x1, 29); x1 ^= x0;
    x0 += x1; x1 = rotl32(x1, 16); x1 ^= x0;
    x0 += x1; x1 = rotl32(x1, 24); x1 ^= x0;
    x0 += k1; x1 += ks2 + 4u;
    x0 += x1; x1 = rotl32(x1, 13); x1 ^= x0;
    x0 += x1; x1 = rotl32(x1, 15); x1 ^= x0;
    x0 += x1; x1 = rotl32(x1, 26); x1 ^= x0;
    x0 += x1; x1 = rotl32(x1,  6); x1 ^= x0;
    x0 += ks2; x1 += k0 + 5u;

    uint32_t bits = lo ? x0 : x1;
    float u = __uint_as_float((bits >> 9) | 0x3F800000u) - 1.0f;  // [0,1)
    const float TINY = 1.17549435e-38f;
    u = u * (1.0f - TINY) + TINY;
    u = fmaxf(u, TINY);
    return -logf(-logf(u));
}

// ---------------------------------------------------------------------------
// Streaming row scan with per-wave double-buffered async global->LDS staging.
// Wave w owns stage[w*256 .. w*256+256): two 128-float buffers.  Each wave
// issues one global_load_async_to_lds_b128 per tile (32 lanes x 16B = 512B)
// and consumes only its own slice -> no cross-wave barriers; synchronization
// purely via the ASYNCcnt split counter (s_wait_asynccnt).
// ---------------------------------------------------------------------------
template <class F>
__device__ __forceinline__ void scan_row(const float* row, float* stage,
                                         int wave, int lane, F f) {
    const int NT = NMAIN / TILE;                     // 148 tiles
    const uint32_t lds_base = (uint32_t)(uintptr_t)(stage + wave * 256);

    {
        const float* g0 = row + /*T=0*/ wave * 128 + lane * 4;
        uint32_t l0 = lds_base + (uint32_t)(lane * 16);
        asm volatile("global_load_async_to_lds_b128 %0, %1, off"
                     :: "v"(l0), "v"(g0) : "memory");
    }
    for (int T = 0; T < NT; ++T) {
        const int p = T & 1;
        if (T + 1 < NT) {
            const float* g = row + (T + 1) * TILE + wave * 128 + lane * 4;
            uint32_t l = lds_base + (uint32_t)(((T + 1) & 1) * 512 + lane * 16);
            asm volatile("global_load_async_to_lds_b128 %0, %1, off"
                         :: "v"(l), "v"(g) : "memory");
            asm volatile("s_wait_asynccnt 0x1" ::: "memory"); // tile T landed
        } else {
            asm volatile("s_wait_asynccnt 0x0" ::: "memory");
        }
        const float4 vv =
            *reinterpret_cast<const float4*>(stage + wave * 256 + p * 128 + lane * 4);
        const int e = T * TILE + wave * 128 + lane * 4;
        f(vv.x, e + 0);
        f(vv.y, e + 1);
        f(vv.z, e + 2);
        f(vv.w, e + 3);
    }
    // 384-element tail: direct global loads
    for (int idx = NMAIN + wave * 32 + lane; idx < VOCAB; idx += NTHREADS)
        f(row[idx], idx);
}

// ---------------------------------------------------------------------------
// Kernel 1: zero penalty bitmask
// ---------------------------------------------------------------------------
__global__ void zero_mask_kernel(uint32_t* mask, int n) {
    for (int i = blockIdx.x * blockDim.x + threadIdx.x; i < n;
         i += gridDim.x * blockDim.x)
        mask[i] = 0u;
}

// ---------------------------------------------------------------------------
// Kernel 2: scatter history token ids into bitmask.  JAX's scatter-set writes
// penalty values computed from the *original* scores, so duplicate ids write
// identical values -> a one-bit "penalized" flag is exact.
// ---------------------------------------------------------------------------
__global__ void scatter_mask_kernel(const int* __restrict__ ids,
                                    uint32_t* __restrict__ mask) {
    int i = blockIdx.x * blockDim.x + threadIdx.x;
    if (i >= BATCH * HIST) return;
    int b = i / HIST;
    int tok = ids[i];
    if (tok >= 0 && tok < VOCAB)
        atomicOr(&mask[(size_t)b * MASKW + (tok >> 5)], 1u << (tok & 31));
}

// ---------------------------------------------------------------------------
// Kernel 3: per-row top-k select + top-p filter + gumbel argmax sample.
// One 256-thread block per batch row.
// ---------------------------------------------------------------------------
__global__ void __launch_bounds__(NTHREADS)
sample_kernel(const float* __restrict__ logits,
              const uint32_t* __restrict__ mask,
              float* __restrict__ out) {
    __shared__ uint32_t hist[HBINS];                  // 16 KB
    __shared__ __align__(16) float stage[2048];       // 8 KB async staging
    __shared__ uint32_t ckey[CAP];                    // 8 KB candidate keys
    __shared__ uint32_t cidx[CAP];                    // 8 KB candidate indices
    __shared__ uint32_t csum[NTHREADS];               // 1 KB chunk sums (16 bins each)
    __shared__ uint32_t gsum[32];                     // group sums (128 bins each)
    __shared__ uint32_t ccount;
    __shared__ uint32_t tbin_s;

    const int b    = blockIdx.x;
    const int tid  = threadIdx.x;
    const int wave = tid >> 5;
    const int lane = tid & 31;
    const float*    row  = logits + (size_t)b * VOCAB;
    const uint32_t* mrow = mask + (size_t)b * MASKW;

    // ---- Pass A: 12-bit key histogram (penalty applied on the fly) ----
    for (int i = tid; i < HBINS; i += NTHREADS) hist[i] = 0u;
    __syncthreads();

    scan_row(row, stage, wave, lane, [&](float x, int idx) {
        uint32_t k = score_key(x, mrow, idx);
        atomicAdd(&hist[k >> 20], 1u);
    });
    __syncthreads();

    // ---- hierarchical suffix-count to find the bin containing rank TOPK ---
    {
        uint32_t s = 0;
        const int base = tid * 16;
        #pragma unroll
        for (int q = 0; q < 16; ++q) s += hist[base + q];
        csum[tid] = s;
    }
    __syncthreads();
    if (tid < 32) {
        uint32_t g = 0;
        #pragma unroll
        for (int q = 0; q < 8; ++q) g += csum[tid * 8 + q];
        gsum[tid] = g;
    }
    __syncthreads();
    if (tid == 0) {
        uint32_t cum = 0;                 // count strictly above current unit
        int g = 31;
        for (; g > 0; --g) { if (cum + gsum[g] >= TOPK) break; cum += gsum[g]; }
        int c = g * 8 + 7;
        for (; c > g * 8; --c) { if (cum + csum[c] >= TOPK) break; cum += csum[c]; }
        int t = c * 16 + 15;
        for (; t > c * 16; --t) { if (cum + hist[t] >= TOPK) break; cum += hist[t]; }
        tbin_s = (uint32_t)t;
        ccount = 0u;
    }
    // pad candidate slots with minimal keys for the bitonic sort
    for (int i = tid; i < CAP; i += NTHREADS) { ckey[i] = 0u; cidx[i] = 0xFFFFFFFFu; }
    __syncthreads();

    // ---- Pass B: collect all candidates >= lower edge of threshold bin ----
    // (whole scores array fits in the 192 MB L2, so this pass is L2-resident)
    const uint32_t kthr = tbin_s << 20;
    scan_row(row, stage, wave, lane, [&](float x, int idx) {
        uint32_t k = score_key(x, mrow, idx);
        if (k >= kthr) {
            uint32_t p = atomicAdd(&ccount, 1u);
            if (p < CAP) { ckey[p] = k; cidx[p] = (uint32_t)idx; }
        }
    });
    __syncthreads();

    // ---- Bitonic sort (descending), adaptive size >= candidate count ------
    int nc_all = (int)ccount; if (nc_all > CAP) nc_all = CAP;
    int S = 64; while (S < nc_all) S <<= 1;          // 64..2048, all pow2
    for (int k = 2; k <= S; k <<= 1) {
        for (int j = k >> 1; j > 0; j >>= 1) {
            for (int i = tid; i < S; i += NTHREADS) {
                int ixj = i ^ j;
                if (ixj > i) {
                    bool up = ((i & k) == 0);
                    uint32_t a = ckey[i], c = ckey[ixj];
                    bool sw = up ? (a < c) : (a > c);
                    if (sw) {
                        ckey[i] = c; ckey[ixj] = a;
                        uint32_t t2 = cidx[i]; cidx[i] = cidx[ixj]; cidx[ixj] = t2;
                    }
                }
            }
            __syncthreads();
        }
    }

    // ---- top-k ties, top-p tail filter, gumbel argmax (serial, ~50 elems) --
    if (tid == 0) {
        int nc = nc_all;
        if (nc < 1) { out[b] = 0.0f; return; }
        int K = (nc < TOPK) ? nc : TOPK;
        // keep ties equal to the k-th value (reference: scores < thr -> -inf)
        uint32_t kth = ckey[K - 1];
        while (K < nc && K < 512 && ckey[K] == kth) ++K;

        // softmax over kept set (everything else has prob 0)
        float m = key2f(ckey[0]);
        float Ssum = 0.0f;
        for (int j2 = 0; j2 < K; ++j2) Ssum += expf(key2f(ckey[j2]) - m);
        // reference removes ascending-cumulative <= 1-TOPP; largest always kept
        // tail_i = sum_{j>=i} p_j ; survivor iff i==0 or tail_i > (1-TOPP)
        const float lim = (1.0f - TOPP) * Ssum;
        float tail = 0.0f; int Ksv = 1;
        for (int j2 = K - 1; j2 >= 1; --j2) {
            tail += expf(key2f(ckey[j2]) - m);
            if (tail > lim) { Ksv = j2 + 1; break; }
        }
        // argmax(score + gumbel) over survivors == softmax-gumbel sample;
        // filtered tokens have prob 0 -> log p = -inf -> can never win.
        float best = -3.4e38f; uint32_t btok = cidx[0];
        for (int i = 0; i < Ksv; ++i) {
            uint32_t tok = cidx[i];
            float g = jax_gumbel42((uint32_t)b * (uint32_t)VOCAB + tok);
            float sc = key2f(ckey[i]) + g;
            if (sc > best) { best = sc; btok = tok; }
        }
        out[b] = (float)btok;
    }
}

// ---------------------------------------------------------------------------
extern "C" void kernel_launch(void* const* d_in, const int* in_sizes, int n_in,
                              void* d_out, int out_size, void* d_ws, size_t ws_size,
                              hipStream_t stream) {
    (void)in_sizes; (void)n_in; (void)out_size; (void)ws_size;
    const int*   ids    = (const int*)d_in[0];     // all_input_ids (B,HIST)
    const float* logits = (const float*)d_in[1];   // (B,1,V) float32
    float*       out    = (float*)d_out;           // B sampled token ids
    uint32_t*    mask   = (uint32_t*)d_ws;         // B*MASKW words (~2.4 MB)

    const int nmask = BATCH * MASKW;
    zero_mask_kernel<<<(nmask + 255) / 256, 256, 0, stream>>>(mask, nmask);
    scatter_mask_kernel<<<(BATCH * HIST + 255) / 256, 256, 0, stream>>>(ids, mask);
    sample_kernel<<<BATCH, NTHREADS, 0, stream>>>(logits, mask, out);
}